// bb_creation_v1_29643864277039
// MI455X (gfx1250) — compile-verified
//
#include <hip/hip_runtime.h>
#include <hip/hip_bf16.h>

// Bounding-box decode + class broadcast for MI455X (gfx1250, wave32).
//
// Output [80, 1, 32768, 85] f32 = 891 MB  -> pure store-bandwidth problem
// (floor ~38us @ 23.3 TB/s). No GEMM exists in this op, so WMMA is
// inapplicable; the CDNA5-specific lever is the cache-control path:
// non-temporal (TH=NT) 16B stores keep the write-once 891MB stream out of
// the 192MB L2, which instead stays resident with the 80x-reused 10.6MB
// class tensor and the 40MB regr tensor.

constexpr unsigned R    = 32768;          // rois
constexpr unsigned CM1  = 80;             // classes - 1
constexpr unsigned NC   = 81;             // class channels
constexpr unsigned ROW  = NC + 4;         // 85 output channels per (c, r)
constexpr unsigned PER_C = R * ROW;       // elements per class slab
constexpr float    Wdim = 640.0f;         // b.shape[2]
constexpr float    Hdim = 640.0f;         // b.shape[1]

typedef float __attribute__((ext_vector_type(4))) f4;

__device__ __forceinline__ float box_component(const float* __restrict__ regr,
                                               const float* __restrict__ rois,
                                               unsigned r, unsigned c, unsigned comp) {
    // rois row: 16B aligned (r*16B); regr row (r*80+c)*4 floats: 16B aligned.
    const f4 ro = *(const f4*)(rois + (size_t)r * 4u);
    const f4 tg = *(const f4*)(regr + ((size_t)r * CM1 + c) * 4u);
    const float x = ro.x, y = ro.y, w = ro.z, h = ro.w;
    // jnp.round == round-half-to-even == v_rndne_f32 (default RNE mode)
    const float gx = __builtin_rintf(w * tg.x + x);
    const float gy = __builtin_rintf(h * tg.y + y);
    const float gw = __builtin_rintf(w * __builtin_expf(tg.z));
    const float gh = __builtin_rintf(h * __builtin_expf(tg.w));
    switch (comp) {
        case 0:  return gx / Wdim;
        case 1:  return gy / Hdim;
        case 2:  return (gx + gw) / Wdim;
        default: return (gy + gh) / Hdim;
    }
}

__global__ __launch_bounds__(256) void bbox_bcast_kernel(
    const float* __restrict__ regr,   // [R, 80, 4]
    const float* __restrict__ cls,    // [R, 81]
    const float* __restrict__ rois,   // [R, 4]
    float*       __restrict__ out,    // [80, 1, R, 85] flat
    unsigned nquads)
{
    const unsigned q = blockIdx.x * blockDim.x + threadIdx.x;
    if (q >= nquads) return;

    const unsigned i   = q * 4u;                  // flat element index (< 2^28)
    const unsigned c   = i / PER_C;               // const divisor -> mulhi+shift
    const unsigned rem = i - c * PER_C;
    const unsigned r   = rem / ROW;               // const divisor -> mulhi+shift
    const unsigned k   = rem - r * ROW;

    f4 v;
    if (k <= NC - 4u) {
        // Fast path (~92% of quads): 4 consecutive class channels of one row.
        // Regular-temporal loads: class tensor is reused 80x -> keep in L2.
        const float* __restrict__ p = cls + (size_t)r * NC + k;
        v.x = p[0]; v.y = p[1]; v.z = p[2]; v.w = p[3];
    } else {
        // Slow path: quad touches box channels and/or a row/class boundary.
        unsigned cc = c, rr = r, kk = k;
        float tmp[4];
#pragma unroll
        for (int j = 0; j < 4; ++j) {
            if (kk < NC) {
                tmp[j] = cls[(size_t)rr * NC + kk];
            } else {
                tmp[j] = box_component(regr, rois, rr, cc, kk - NC);
            }
            if (++kk == ROW) { kk = 0; if (++rr == R) { rr = 0; ++cc; } }
        }
        v.x = tmp[0]; v.y = tmp[1]; v.z = tmp[2]; v.w = tmp[3];
    }

    // Write-once 891MB stream: non-temporal 16B store (global_store_b128,
    // TH=NT) -> full cachelines, no L2 pollution.
    __builtin_nontemporal_store(v, (f4*)(out + (size_t)i));
}

extern "C" void kernel_launch(void* const* d_in, const int* in_sizes, int n_in,
                              void* d_out, int out_size, void* d_ws, size_t ws_size,
                              hipStream_t stream) {
    (void)in_sizes; (void)n_in; (void)d_ws; (void)ws_size;
    const float* regr = (const float*)d_in[0];   // [1, R, 320]
    const float* cls  = (const float*)d_in[1];   // [1, R, 81]
    const float* rois = (const float*)d_in[2];   // [1, R, 4]
    // d_in[3] = b: only its static 640x640 shape is used (baked in).
    float* out = (float*)d_out;                  // [80, 1, R, 85]

    const unsigned nquads = (unsigned)(out_size / 4);  // out_size % 4 == 0
    const unsigned threads = 256;                       // 8 wave32 waves/block
    const unsigned blocks  = (nquads + threads - 1) / threads;
    bbox_bcast_kernel<<<blocks, threads, 0, stream>>>(regr, cls, rois, out, nquads);
}